// JointLocationLoss_8091718386186
// MI455X (gfx1250) — compile-verified
//
#include <hip/hip_runtime.h>
#include <hip/hip_bf16.h>
#include <stdint.h>
#include <math.h>

#if __has_builtin(__builtin_amdgcn_exp2f)
__device__ __forceinline__ float fast_exp2(float x) { return __builtin_amdgcn_exp2f(x); }
#else
__device__ __forceinline__ float fast_exp2(float x) { return exp2f(x); }
#endif

static constexpr int   kJ      = 21;             // joints
static constexpr int   kD      = 64, kH = 64, kW = 64;
static constexpr int   kNElem  = kD * kH * kW;   // 262144 per (b,j)
static constexpr int   kNF4    = kNElem / 4;     // 65536 float4s
static constexpr int   kTPB    = 256;            // 8 wave32
static constexpr int   kF4PerT = kNF4 / kTPB;    // 256 float4s per thread
static constexpr int   kTiles  = kF4PerT / 4;    // 64 tiles of 4 float4s/thread (16 KB/tile)
static constexpr float kLog2e  = 1.4426950408889634f;

__global__ void JLL_zero_out(float* out) { out[0] = 0.0f; }

__global__ __launch_bounds__(kTPB)
void JLL_softargmax_l1_kernel(const float* __restrict__ hm,
                              const float* __restrict__ gt_coord,
                              const float* __restrict__ gt_vis,
                              float* __restrict__ out,
                              int Bnum)
{
    // 4-deep async staging: 4 bufs x 4 float4-rows x 256 threads x 16B = 64 KB
    __shared__ float4 sbuf[4][4][kTPB];
    __shared__ float rm[kTPB], rs[kTPB], rx[kTPB], ry[kTPB], rz[kTPB];

    const int blk = blockIdx.x;            // b * kJ + j
    const int b   = blk / kJ;
    const int j   = blk - b * kJ;
    const float4* __restrict__ src =
        reinterpret_cast<const float4*>(hm + (size_t)blk * (size_t)kNElem);
    const int t = threadIdx.x;

    // Issue one tile (4 x b128 per lane) as async DMA into this lane's LDS slots.
    auto issue_tile = [&](int tile) {
        const int buf = tile & 3;
        #pragma unroll
        for (int k4 = 0; k4 < 4; ++k4) {
            const float4* g = src + t + kTPB * (4 * tile + k4);
            // wave-relative LDS byte address (generic LDS ptr low 32 bits)
            const uint32_t l = (uint32_t)(uintptr_t)(&sbuf[buf][k4][t]);
            asm volatile("global_load_async_to_lds_b128 %0, %1, off"
                         :: "v"(l), "v"(g) : "memory");
        }
    };

    // Prime the pipeline 3 tiles deep (12 outstanding async loads per wave).
    issue_tile(0);
    issue_tile(1);
    issue_tile(2);

    // Online (tile-granular) softmax accumulators, per thread.
    float  m  = -3.0e38f;
    float4 s4 = make_float4(0.f, 0.f, 0.f, 0.f);
    float  sy = 0.f, sz = 0.f;

    for (int tile = 0; tile < kTiles; ++tile) {
        if (tile + 3 < kTiles) {
            issue_tile(tile + 3);
            if (tile + 16 < kTiles) {
                // warm L2 well ahead of the async pipeline (global_prefetch_b8)
                __builtin_prefetch((const void*)&src[t + kTPB * 4 * (tile + 16)], 0, 3);
            }
            asm volatile("s_wait_asynccnt 0xc" ::: "memory");   // oldest tile landed
        } else {
            const int ahead = kTiles - 1 - tile;                // tiles still in flight past this one
            if (ahead == 2)      asm volatile("s_wait_asynccnt 0x8" ::: "memory");
            else if (ahead == 1) asm volatile("s_wait_asynccnt 0x4" ::: "memory");
            else                 asm volatile("s_wait_asynccnt 0x0" ::: "memory");
        }

        const int buf = tile & 3;
        const float4 r0 = sbuf[buf][0][t];
        const float4 r1 = sbuf[buf][1][t];
        const float4 r2 = sbuf[buf][2][t];
        const float4 r3 = sbuf[buf][3][t];

        // tile max (16 values)
        const float cm =
            fmaxf(fmaxf(fmaxf(fmaxf(r0.x, r0.y), fmaxf(r0.z, r0.w)),
                        fmaxf(fmaxf(r1.x, r1.y), fmaxf(r1.z, r1.w))),
                  fmaxf(fmaxf(fmaxf(r2.x, r2.y), fmaxf(r2.z, r2.w)),
                        fmaxf(fmaxf(r3.x, r3.y), fmaxf(r3.z, r3.w))));

        const float mn = fmaxf(m, cm);
        const float sc = fast_exp2((m - mn) * kLog2e);  // ==1 when max unchanged
        m = mn;
        s4.x *= sc; s4.y *= sc; s4.z *= sc; s4.w *= sc;
        sy *= sc;  sz *= sc;

        const float mneg = -m * kLog2e;
        const float4 rr[4] = { r0, r1, r2, r3 };
        #pragma unroll
        for (int k4 = 0; k4 < 4; ++k4) {
            const float4 r = rr[k4];
            const int k  = 4 * tile + k4;         // float4 row index for this thread
            const int hd = (t >> 4) + 16 * k;     // == element_idx >> 6 (shared by all 4 comps)
            const float hf = (float)(hd & 63);    // y index
            const float df = (float)(hd >> 6);    // z index
            const float e0 = fast_exp2(fmaf(r.x, kLog2e, mneg));
            const float e1 = fast_exp2(fmaf(r.y, kLog2e, mneg));
            const float e2 = fast_exp2(fmaf(r.z, kLog2e, mneg));
            const float e3 = fast_exp2(fmaf(r.w, kLog2e, mneg));
            s4.x += e0; s4.y += e1; s4.z += e2; s4.w += e3;
            const float es = (e0 + e1) + (e2 + e3);
            sy = fmaf(es, hf, sy);
            sz = fmaf(es, df, sz);
        }
    }

    // x index is constant per (thread, component): w_c = (4t + c) & 63 = w0 + c
    const float w0 = (float)((4 * t) & 63);
    const float sx = s4.x * w0 + s4.y * (w0 + 1.f) + s4.z * (w0 + 2.f) + s4.w * (w0 + 3.f);
    const float s  = (s4.x + s4.y) + (s4.z + s4.w);

    // Cross-thread merge with max-rescaling (LDS tree over 256 threads)
    rm[t] = m; rs[t] = s; rx[t] = sx; ry[t] = sy; rz[t] = sz;
    __syncthreads();
    for (int off = kTPB / 2; off > 0; off >>= 1) {
        if (t < off) {
            const float m1 = rm[t], m2 = rm[t + off];
            const float M  = fmaxf(m1, m2);
            const float a1 = fast_exp2((m1 - M) * kLog2e);
            const float a2 = fast_exp2((m2 - M) * kLog2e);
            rm[t] = M;
            rs[t] = rs[t] * a1 + rs[t + off] * a2;
            rx[t] = rx[t] * a1 + rx[t + off] * a2;
            ry[t] = ry[t] * a1 + ry[t + off] * a2;
            rz[t] = rz[t] * a1 + rz[t + off] * a2;
        }
        __syncthreads();
    }

    if (t == 0) {
        const float S   = rs[0];
        const float inv = 1.0f / (S * 64.0f);        // /W (==H==D)
        const float x = rx[0] * inv - 0.5f;
        const float y = ry[0] * inv - 0.5f;
        const float z = rz[0] * inv - 0.5f;
        const int base = b * (kJ * 3) + j * 3;
        const float l = fabsf(x - gt_coord[base + 0]) * gt_vis[base + 0]
                      + fabsf(y - gt_coord[base + 1]) * gt_vis[base + 1]
                      + fabsf(z - gt_coord[base + 2]) * gt_vis[base + 2];
        atomicAdd(out, l * (1.0f / (float)Bnum));
    }
}

extern "C" void kernel_launch(void* const* d_in, const int* in_sizes, int n_in,
                              void* d_out, int out_size, void* d_ws, size_t ws_size,
                              hipStream_t stream) {
    const float* hm  = (const float*)d_in[0];   // [B, J*D, H, W] fp32
    const float* gtc = (const float*)d_in[1];   // [B, J*3]
    const float* gtv = (const float*)d_in[2];   // [B, J*3]
    float* out = (float*)d_out;                 // scalar loss

    const int B = in_sizes[1] / (kJ * 3);       // 64 for the reference setup

    JLL_zero_out<<<1, 1, 0, stream>>>(out);
    JLL_softargmax_l1_kernel<<<B * kJ, kTPB, 0, stream>>>(hm, gtc, gtv, out, B);
}